// LongTermMemory_12103217840695
// MI455X (gfx1250) — compile-verified
//
#include <hip/hip_runtime.h>

// MI455X / gfx1250, wave32. f32 WMMA GEMM (16x16x4) with double-buffered LDS
// slabs filled by GLOBAL_LOAD_ASYNC_TO_LDS_B128 (ASYNCcnt), + argmax + gather.

typedef __attribute__((ext_vector_type(2))) float v2f;
typedef __attribute__((ext_vector_type(8))) float v8f;

#define DDIM   512
#define BROWS  512
#define MROWS  65536
#define KS     32                  // k-slab width staged in LDS
#define LDA    36                  // padded LDS row stride (floats)
#define CTILE  128                 // ctx rows per block
#define MSTRIP 256                 // mem rows per block (4 strips of 64)
#define MITER  4
#define NSLAB  (DDIM / KS)         // 16
#define NSTAGE (MITER * NSLAB)     // 64
#define NCHUNK (MROWS / MSTRIP)    // 256

static __device__ __forceinline__ v8f wmma_f32(v2f a, v2f b, v8f c) {
  return __builtin_amdgcn_wmma_f32_16x16x4_f32(false, a, false, b, (short)0, c,
                                               false, false);
}

static __device__ __forceinline__ void async_b128(const float* gptr, void* lptr) {
  const uint32_t l = (uint32_t)(uintptr_t)lptr;      // flat addr low 32 = LDS offset
  asm volatile("global_load_async_to_lds_b128 %0, %1, off"
               :: "v"(l), "v"((uint64_t)(uintptr_t)gptr) : "memory");
}

static __device__ __forceinline__ void wait_async0() {
#if __has_builtin(__builtin_amdgcn_s_wait_asynccnt)
  __builtin_amdgcn_s_wait_asynccnt(0);
#else
  asm volatile("s_wait_asynccnt 0x0" ::: "memory");
#endif
}

// ---------------------------------------------------------------------------
// Kernel 1: inv_norm[m] = 1/sqrt(max(sum(memory[m]^2), 1e-12))
// ---------------------------------------------------------------------------
__global__ __launch_bounds__(256) void ltm_norms(const float* __restrict__ mem,
                                                 float* __restrict__ inv) {
  const int wave = threadIdx.x >> 5;
  const int lane = threadIdx.x & 31;
  const int row  = blockIdx.x * 8 + wave;
  const float4* m4 = (const float4*)(mem + (size_t)row * DDIM);
  float s = 0.0f;
#pragma unroll
  for (int i = 0; i < 4; ++i) {
    float4 v = m4[i * 32 + lane];
    s += v.x * v.x + v.y * v.y + v.z * v.z + v.w * v.w;
  }
#pragma unroll
  for (int m = 16; m >= 1; m >>= 1) s += __shfl_xor(s, m, 32);
  if (lane == 0) inv[row] = 1.0f / sqrtf(fmaxf(s, 1e-12f));
}

// ---------------------------------------------------------------------------
// Kernel 2: block = 128 ctx rows x 64 mem cols, 8 waves (4 ctx-groups x 2
// mem-groups), each wave a 32x32 tile (2x2 WMMA frags). K in 32-wide slabs,
// async-DMA'd into double-buffered LDS; one barrier per stage.
// ---------------------------------------------------------------------------
__global__ __launch_bounds__(256) void ltm_simarg(const float* __restrict__ ctx,
                                                  const float* __restrict__ mem,
                                                  const float* __restrict__ inv,
                                                  float* __restrict__ pval,
                                                  int*   __restrict__ pidx) {
  __shared__ float lsA[2][CTILE * LDA];   // 128 rows x 32 k (padded)
  __shared__ float lsB[2][64 * LDA];      // 64 rows x 32 k (padded)
  __shared__ float rvalS[8][32];
  __shared__ int   ridxS[8][32];

  const int tid  = threadIdx.x;
  const int wave = tid >> 5;
  const int lane = tid & 31;
  const int nl   = lane & 15;
  const int h    = lane >> 4;     // half -> k-pair offset 2h, C-row offset 8h
  const int wy   = wave >> 1;     // ctx group 0..3 (32 rows each)
  const int wx   = wave & 1;      // mem group 0..1 (32 cols each)

  const int ctx_base   = blockIdx.x * CTILE;
  const int strip_base = blockIdx.y * MSTRIP;

  const int aoff0 = (wy * 32 + nl) * LDA + 2 * h;
  const int aoff1 = aoff0 + 16 * LDA;
  const int boff0 = (wx * 32 + nl) * LDA + 2 * h;
  const int boff1 = boff0 + 16 * LDA;

  float bval[2][8];
  int   bidx[2][8];
#pragma unroll
  for (int sy = 0; sy < 2; ++sy)
#pragma unroll
    for (int r = 0; r < 8; ++r) { bval[sy][r] = -3.4e38f; bidx[sy][r] = 0; }

  v8f acc[2][2];

  // Issue this stage's 6 per-thread async b128 DMAs (4 A-tile + 2 B-tile).
  auto issue_async = [&](int g, int buf) {
    const int it = g >> 4;           // which 64-col strip
    const int s  = g & (NSLAB - 1);  // which k-slab
    const int kb = s * KS;
    const int mb = strip_base + it * 64;
#pragma unroll
    for (int i = 0; i < 4; ++i) {    // A: 1024 float4 / 256 threads
      const int f = i * 256 + tid;
      const int r = f >> 3, c = f & 7;
      async_b128(ctx + (size_t)(ctx_base + r) * DDIM + kb + c * 4,
                 &lsA[buf][r * LDA + c * 4]);
    }
#pragma unroll
    for (int i = 0; i < 2; ++i) {    // B: 512 float4 / 256 threads
      const int f = i * 256 + tid;
      const int r = f >> 3, c = f & 7;
      async_b128(mem + (size_t)(mb + r) * DDIM + kb + c * 4,
                 &lsB[buf][r * LDA + c * 4]);
    }
  };

  issue_async(0, 0);
  wait_async0();
  __syncthreads();

  for (int g = 0; g < NSTAGE; ++g) {
    const int buf = g & 1;
    const int it  = g >> 4;
    const int s   = g & (NSLAB - 1);

    // Prefetch next slab into the idle buffer (its readers finished before
    // the previous barrier; DMA overlaps the WMMA compute below).
    if (g + 1 < NSTAGE) issue_async(g + 1, (g + 1) & 1);

    if (s == 0) {
#pragma unroll
      for (int sy = 0; sy < 2; ++sy)
#pragma unroll
        for (int sx = 0; sx < 2; ++sx) acc[sy][sx] = (v8f){};
    }

    const float* pa0 = &lsA[buf][aoff0];
    const float* pa1 = &lsA[buf][aoff1];
    const float* pb0 = &lsB[buf][boff0];
    const float* pb1 = &lsB[buf][boff1];
#pragma unroll
    for (int k = 0; k < KS; k += 4) {
      v2f a0 = *(const v2f*)(pa0 + k);
      v2f a1 = *(const v2f*)(pa1 + k);
      v2f b0 = *(const v2f*)(pb0 + k);
      v2f b1 = *(const v2f*)(pb1 + k);
      acc[0][0] = wmma_f32(a0, b0, acc[0][0]);
      acc[0][1] = wmma_f32(a0, b1, acc[0][1]);
      acc[1][0] = wmma_f32(a1, b0, acc[1][0]);
      acc[1][1] = wmma_f32(a1, b1, acc[1][1]);
    }

    if (s == NSLAB - 1) {   // end of a 64-col strip: scale + fold argmax
      const int mb = strip_base + it * 64;
#pragma unroll
      for (int sx = 0; sx < 2; ++sx) {
        const int midx = mb + wx * 32 + sx * 16 + nl;
        const float sc = inv[midx];
#pragma unroll
        for (int sy = 0; sy < 2; ++sy)
#pragma unroll
          for (int r = 0; r < 8; ++r) {
            const float v = acc[sy][sx][r] * sc;
            if (v > bval[sy][r]) { bval[sy][r] = v; bidx[sy][r] = midx; }
          }
      }
    }

    wait_async0();      // next buffer fully landed in LDS
    __syncthreads();    // all waves done (their DS reads retired pre-signal)
  }

  // ---- cross-lane reduce over the 16 cols (nl) each half covers ----
#pragma unroll
  for (int sy = 0; sy < 2; ++sy)
#pragma unroll
    for (int r = 0; r < 8; ++r)
#pragma unroll
      for (int m = 8; m >= 1; m >>= 1) {
        float ov = __shfl_xor(bval[sy][r], m, 32);
        int   oi = __shfl_xor(bidx[sy][r], m, 32);
        if (ov > bval[sy][r] || (ov == bval[sy][r] && oi < bidx[sy][r])) {
          bval[sy][r] = ov; bidx[sy][r] = oi;
        }
      }
  if (nl == 0) {
#pragma unroll
    for (int sy = 0; sy < 2; ++sy)
#pragma unroll
      for (int r = 0; r < 8; ++r) {
        rvalS[wave][sy * 16 + 8 * h + r] = bval[sy][r];
        ridxS[wave][sy * 16 + 8 * h + r] = bidx[sy][r];
      }
  }
  __syncthreads();

  if (tid < CTILE) {
    const int row  = tid;          // 0..127
    const int wgrp = row >> 5;
    const int lrow = row & 31;
    float v0 = rvalS[wgrp * 2 + 0][lrow];
    int   i0 = ridxS[wgrp * 2 + 0][lrow];
    const float v1 = rvalS[wgrp * 2 + 1][lrow];
    const int   i1 = ridxS[wgrp * 2 + 1][lrow];
    if (v1 > v0 || (v1 == v0 && i1 < i0)) { v0 = v1; i0 = i1; }
    const int o = (blockIdx.x * NCHUNK + blockIdx.y) * CTILE + row;
    pval[o] = v0;
    pidx[o] = i0;
  }
}

// ---------------------------------------------------------------------------
// Kernel 3: one block per ctx row; reduce 256 chunk-partials (lowest index
// wins ties, matching argmax-first), then gather memory[best].
// ---------------------------------------------------------------------------
__global__ __launch_bounds__(256) void ltm_reduce_gather(const float* __restrict__ mem,
                                                         const float* __restrict__ pval,
                                                         const int*   __restrict__ pidx,
                                                         float* __restrict__ out) {
  __shared__ float sv[8];
  __shared__ int   si[8];
  __shared__ int   sbest;

  const int b    = blockIdx.x;        // ctx row 0..511
  const int tile = b >> 7;
  const int row  = b & (CTILE - 1);
  const int tid  = threadIdx.x;
  const int wave = tid >> 5;
  const int lane = tid & 31;

  const int o = (tile * NCHUNK + tid) * CTILE + row;   // chunk = tid (256)
  float bv = pval[o];
  int   bi = pidx[o];
#pragma unroll
  for (int m = 16; m >= 1; m >>= 1) {
    float ov = __shfl_xor(bv, m, 32);
    int   oi = __shfl_xor(bi, m, 32);
    if (ov > bv || (ov == bv && oi < bi)) { bv = ov; bi = oi; }
  }
  if (lane == 0) { sv[wave] = bv; si[wave] = bi; }
  __syncthreads();
  if (tid == 0) {
    float v = sv[0]; int i = si[0];
#pragma unroll
    for (int w = 1; w < 8; ++w) {
      if (sv[w] > v || (sv[w] == v && si[w] < i)) { v = sv[w]; i = si[w]; }
    }
    sbest = i;
  }
  __syncthreads();

  const int best = sbest;
  const float2* src = (const float2*)(mem + (size_t)best * DDIM);
  float2*       dst = (float2*)(out + (size_t)b * DDIM);
  dst[tid] = src[tid];
}

// ---------------------------------------------------------------------------
extern "C" void kernel_launch(void* const* d_in, const int* in_sizes, int n_in,
                              void* d_out, int out_size, void* d_ws, size_t ws_size,
                              hipStream_t stream) {
  const float* ctx = (const float*)d_in[0];   // [512, 512]
  const float* mem = (const float*)d_in[1];   // [65536, 512]
  float* out = (float*)d_out;                 // [1, 512, 512]

  float* inv  = (float*)d_ws;                         // 65536 f
  float* pval = inv + MROWS;                          // 4*256*128 f
  int*   pidx = (int*)(pval + (BROWS / CTILE) * NCHUNK * CTILE);

  ltm_norms<<<MROWS / 8, 256, 0, stream>>>(mem, inv);
  ltm_simarg<<<dim3(BROWS / CTILE, NCHUNK), 256, 0, stream>>>(ctx, mem, inv,
                                                             pval, pidx);
  ltm_reduce_gather<<<BROWS, 256, 0, stream>>>(mem, pval, pidx, out);
}